// BlockWithAttention_85925115724369
// MI455X (gfx1250) — compile-verified
//
#include <hip/hip_runtime.h>
#include <hip/hip_bf16.h>

typedef __attribute__((ext_vector_type(2))) float v2f;
typedef __attribute__((ext_vector_type(8))) float v8f;

// Padded pair-row strides (in float2 units). 2*PLD % 64 == 32 so the upper
// half-wave (kpair+1) hits the other 32 LDS banks -> conflict-free b64 loads.
#define PLD32  48
#define PLD64  80
#define PLD128 144

// ---------------------------------------------------------------------------
// D = A(16x4) * B(4x16) + C, exact fp32 on the CDNA5 matrix pipe.
// ---------------------------------------------------------------------------
__device__ __forceinline__ v8f wmma_f32(v2f a, v2f b, v8f c) {
    return __builtin_amdgcn_wmma_f32_16x16x4_f32(
        /*neg_a=*/false, a, /*neg_b=*/false, b,
        /*c_mod=*/(short)0, c, /*reuse_a=*/false, /*reuse_b=*/false);
}

// K-paired LDS tile: element (k, col) lives at base[((k>>1)*pld + col)*2 + (k&1)].
// A fragment (two consecutive k at one row/col) is a single aligned ds_load_b64.
__device__ __forceinline__ v2f ldfrag2(const float* base, int lane, int off, int kb, int pld) {
    const int kpair = (kb >> 1) + (lane >> 4);   // kb is a multiple of 4
    const int r = lane & 15;
    const float2 v = *(const float2*)(base + (kpair * pld + off + r) * 2);
    v2f f; f.x = v.x; f.y = v.y;
    return f;
}
__device__ __forceinline__ void stpair(float* base, int pld, int k /*even*/, int col,
                                       float a, float b) {
    *(float2*)(base + ((k >> 1) * pld + col) * 2) = make_float2(a, b);
}

// ---------------------------------------------------------------------------
__global__ void zero_kernel(float* p, int n) {
    int i = blockIdx.x * 256 + threadIdx.x;
    if (i < n) p[i] = 0.0f;
}

// ---------------------------------------------------------------------------
// Time MLP: te = relu(relu(t @ w1^T + b1) @ w2^T + b2).  B=16, T=256. Tiny.
// ---------------------------------------------------------------------------
__global__ __launch_bounds__(256) void time_mlp_kernel(
    const float* __restrict__ t, const float* __restrict__ w1, const float* __restrict__ b1,
    const float* __restrict__ w2, const float* __restrict__ b2, float* __restrict__ te) {
    __shared__ float st[16 * 256];
    __shared__ float sh[16 * 256];
    const int tid = threadIdx.x;
    for (int i = tid; i < 4096; i += 256) st[i] = t[i];
    __syncthreads();
    for (int b = 0; b < 16; ++b) {
        float acc = b1[tid];
        for (int k = 0; k < 256; ++k) acc += st[b * 256 + k] * w1[tid * 256 + k];
        sh[b * 256 + tid] = fmaxf(acc, 0.0f);
    }
    __syncthreads();
    for (int b = 0; b < 16; ++b) {
        float acc = b2[tid];
        for (int k = 0; k < 256; ++k) acc += sh[b * 256 + k] * w2[tid * 256 + k];
        te[b * 256 + tid] = fmaxf(acc, 0.0f);
    }
}

// ---------------------------------------------------------------------------
// 3x3 SAME conv as implicit GEMM: M=256 (oc), N=16384 (B*H*W), K=2304 (Cin*9).
// Block 128 thr (4 waves), block tile 64x128, per-wave 32x64 (8 WMMA / 6 loads).
// Fused +bias, optional ReLU, optional per-channel sum/sumsq for BatchNorm.
// ---------------------------------------------------------------------------
__global__ __launch_bounds__(128) void conv3x3_wmma_kernel(
    const float* __restrict__ X, const float* __restrict__ Wt,
    const float* __restrict__ bias, float* __restrict__ Out,
    float* __restrict__ stats, int doRelu) {
    __shared__ float As[8 * PLD64 * 2];    // [kpair][m:64][2]
    __shared__ float Bs[8 * PLD128 * 2];   // [kpair][n:128][2]
    __shared__ float sSum[64];
    __shared__ float sSq[64];

    const int tid  = threadIdx.x;
    const int lane = tid & 31;
    const int wave = tid >> 5;
    const int wm = wave >> 1, wn = wave & 1;
    const int p0 = blockIdx.x * 128;         // pixel tile base (within one image)
    const int m0 = blockIdx.y * 64;          // out-channel tile base
    const int b  = p0 >> 10;
    const int ybase = (p0 & 1023) >> 5;      // first image row of this tile

    if (stats != nullptr && tid < 64) { sSum[tid] = 0.0f; sSq[tid] = 0.0f; }

    v8f acc[2][4] = {};

    const int am = tid >> 1;                 // A loader: channel row (0..63)
    const int ak = (tid & 1) * 8;            // A loader: k sub-offset
    const int kp = tid >> 4;                 // B loader: k-pair (0..7)
    const int bn0 = (tid & 15) * 8;          // B loader: n sub-offset

    for (int k0 = 0; k0 < 2304; k0 += 16) {
        __syncthreads();
        // --- weight tile: 8 consecutive k for one channel, paired stores ---
        {
            const float* src = Wt + (m0 + am) * 2304 + k0 + ak;
            #pragma unroll
            for (int i = 0; i < 8; i += 4) {
                float4 v = *(const float4*)(src + i);
                stpair(As, PLD64, ak + i,     am, v.x, v.y);
                stpair(As, PLD64, ak + i + 2, am, v.z, v.w);
            }
        }
        // --- im2col tile: each thread fills k-pair (2kp, 2kp+1) for 8 pixels ---
        {
            const int kkA = k0 + 2 * kp;
            const int cA = kkA / 9;      int rsA = kkA - cA * 9;
            const int rA = rsA / 3;      const int sA = rsA - rA * 3;
            const int kkB = kkA + 1;
            const int cB = kkB / 9;      int rsB = kkB - cB * 9;
            const int rB = rsB / 3;      const int sB = rsB - rB * 3;
            const float* xa = X + ((b * 256 + cA) << 10);
            const float* xb = X + ((b * 256 + cB) << 10);
            #pragma unroll
            for (int i = 0; i < 8; ++i) {
                const int n = bn0 + i;
                const int ya = ybase + (n >> 5) + rA - 1, xaa = (n & 31) + sA - 1;
                const int yb = ybase + (n >> 5) + rB - 1, xbb = (n & 31) + sB - 1;
                float va = 0.0f, vb = 0.0f;
                if ((unsigned)ya < 32u && (unsigned)xaa < 32u) va = xa[(ya << 5) + xaa];
                if ((unsigned)yb < 32u && (unsigned)xbb < 32u) vb = xb[(yb << 5) + xbb];
                stpair(Bs, PLD128, 2 * kp, n, va, vb);
            }
        }
        __syncthreads();
        #pragma unroll
        for (int ks = 0; ks < 4; ++ks) {
            const int kb = ks * 4;
            v2f a0 = ldfrag2(As, lane, 32 * wm + 0,  kb, PLD64);
            v2f a1 = ldfrag2(As, lane, 32 * wm + 16, kb, PLD64);
            v2f b0 = ldfrag2(Bs, lane, 64 * wn + 0,  kb, PLD128);
            v2f b1 = ldfrag2(Bs, lane, 64 * wn + 16, kb, PLD128);
            v2f b2 = ldfrag2(Bs, lane, 64 * wn + 32, kb, PLD128);
            v2f b3 = ldfrag2(Bs, lane, 64 * wn + 48, kb, PLD128);
            acc[0][0] = wmma_f32(a0, b0, acc[0][0]);
            acc[0][1] = wmma_f32(a0, b1, acc[0][1]);
            acc[0][2] = wmma_f32(a0, b2, acc[0][2]);
            acc[0][3] = wmma_f32(a0, b3, acc[0][3]);
            acc[1][0] = wmma_f32(a1, b0, acc[1][0]);
            acc[1][1] = wmma_f32(a1, b1, acc[1][1]);
            acc[1][2] = wmma_f32(a1, b2, acc[1][2]);
            acc[1][3] = wmma_f32(a1, b3, acc[1][3]);
        }
    }

    const int kh = lane >> 4, r15 = lane & 15;
    #pragma unroll
    for (int mi = 0; mi < 2; ++mi) {
        #pragma unroll
        for (int ni = 0; ni < 4; ++ni) {
            const int chanLocalBase = 32 * wm + 16 * mi + 8 * kh;
            const int pix = p0 + 64 * wn + 16 * ni + r15;
            const int pin = pix & 1023;
            #pragma unroll
            for (int i = 0; i < 8; ++i) {
                const int chan = m0 + chanLocalBase + i;
                float v = acc[mi][ni][i] + bias[chan];
                if (doRelu) v = fmaxf(v, 0.0f);
                Out[(b << 18) + (chan << 10) + pin] = v;
                if (stats != nullptr) {
                    atomicAdd(&sSum[chanLocalBase + i], v);   // ds_add_f32
                    atomicAdd(&sSq[chanLocalBase + i], v * v);
                }
            }
        }
    }
    if (stats != nullptr) {
        __syncthreads();
        if (tid < 64)        atomicAdd(&stats[m0 + tid], sSum[tid]);
        else if (tid < 128)  atomicAdd(&stats[256 + m0 + tid - 64], sSq[tid - 64]);
    }
}

// ---------------------------------------------------------------------------
// BatchNorm apply (training-mode stats, biased variance) + optional te add.
// ---------------------------------------------------------------------------
__global__ __launch_bounds__(256) void bn_apply_kernel(
    float* __restrict__ buf, const float* __restrict__ stats,
    const float* __restrict__ g, const float* __restrict__ bb,
    const float* __restrict__ te) {
    const int idx = blockIdx.x * 256 + threadIdx.x;
    const int c = (idx >> 10) & 255;
    const int b = idx >> 18;
    const float mean = stats[c] * (1.0f / 16384.0f);
    const float var  = stats[256 + c] * (1.0f / 16384.0f) - mean * mean;
    const float inv  = rsqrtf(var + 1e-5f);
    float v = (buf[idx] - mean) * inv * g[c] + bb[c];
    if (te != nullptr) v += te[b * 256 + c];
    buf[idx] = v;
}

// ---------------------------------------------------------------------------
// 1x1 projection GEMM: Out[o,p] = sum_c W[o,c]*Y[b,c,n] + bias[o].
// Tile 32(M) x 64(N), K=256.  qLayout=1 -> [B,N,O] (Q); else [B,O,N] (K,V).
// ---------------------------------------------------------------------------
__global__ __launch_bounds__(128) void proj1x1_wmma_kernel(
    const float* __restrict__ Y, const float* __restrict__ Wt,
    const float* __restrict__ bias, float* __restrict__ Out, int O, int qLayout) {
    __shared__ float As[8 * PLD32 * 2];
    __shared__ float Bs[8 * PLD64 * 2];
    const int tid = threadIdx.x, lane = tid & 31, wave = tid >> 5;
    const int wm = wave >> 1, wn = wave & 1;
    const int p0 = blockIdx.x * 64;
    const int m0 = blockIdx.y * 32;
    const int b = p0 >> 10, nbase = p0 & 1023;

    v8f acc[2] = {};
    const int am = tid >> 2, ak = (tid & 3) * 4;   // A: 32 rows x 16 k
    const int kp = tid >> 4, bn0 = (tid & 15) * 4; // B: k-pair x 4 pixels

    for (int k0 = 0; k0 < 256; k0 += 16) {
        __syncthreads();
        {
            float4 v = *(const float4*)(Wt + (m0 + am) * 256 + k0 + ak);
            stpair(As, PLD32, ak,     am, v.x, v.y);
            stpair(As, PLD32, ak + 2, am, v.z, v.w);
        }
        {
            const float* s0 = Y + ((b * 256 + k0 + 2 * kp) << 10) + nbase + bn0;
            float4 u = *(const float4*)(s0);
            float4 w = *(const float4*)(s0 + 1024);
            stpair(Bs, PLD64, 2 * kp, bn0 + 0, u.x, w.x);
            stpair(Bs, PLD64, 2 * kp, bn0 + 1, u.y, w.y);
            stpair(Bs, PLD64, 2 * kp, bn0 + 2, u.z, w.z);
            stpair(Bs, PLD64, 2 * kp, bn0 + 3, u.w, w.w);
        }
        __syncthreads();
        #pragma unroll
        for (int ks = 0; ks < 4; ++ks) {
            const int kb = ks * 4;
            v2f a  = ldfrag2(As, lane, 16 * wm, kb, PLD32);
            v2f b0 = ldfrag2(Bs, lane, 32 * wn + 0,  kb, PLD64);
            v2f b1 = ldfrag2(Bs, lane, 32 * wn + 16, kb, PLD64);
            acc[0] = wmma_f32(a, b0, acc[0]);
            acc[1] = wmma_f32(a, b1, acc[1]);
        }
    }
    const int kh = lane >> 4, r15 = lane & 15;
    #pragma unroll
    for (int ni = 0; ni < 2; ++ni) {
        const int n = nbase + 32 * wn + 16 * ni + r15;
        #pragma unroll
        for (int i = 0; i < 8; ++i) {
            const int o = m0 + 16 * wm + 8 * kh + i;
            const float v = acc[ni][i] + bias[o];
            if (qLayout) Out[(b * 1024 + n) * O + o] = v;
            else         Out[(b * O + o) * 1024 + n] = v;
        }
    }
}

// ---------------------------------------------------------------------------
// Attention scores (one batch): S[n,m] = sum_q Q[n,q]*K[q,m].  K=32 resident.
// ---------------------------------------------------------------------------
__global__ __launch_bounds__(128) void scores_wmma_kernel(
    const float* __restrict__ Qb, const float* __restrict__ Kb, float* __restrict__ S) {
    __shared__ float As[16 * PLD64 * 2];   // [qpair][n-row][2]
    __shared__ float Bs[16 * PLD64 * 2];   // [qpair][m-col][2]
    const int tid = threadIdx.x, lane = tid & 31, wave = tid >> 5;
    const int wm = wave >> 1, wn = wave & 1;
    const int row0 = blockIdx.y * 64;   // n
    const int col0 = blockIdx.x * 64;   // m

    {   // Q panel: 16 consecutive q for one row
        const int m = tid >> 1, k16 = (tid & 1) * 16;
        const float* src = Qb + (row0 + m) * 32 + k16;
        #pragma unroll
        for (int i = 0; i < 16; i += 4) {
            float4 v = *(const float4*)(src + i);
            stpair(As, PLD64, k16 + i,     m, v.x, v.y);
            stpair(As, PLD64, k16 + i + 2, m, v.z, v.w);
        }
    }
    {   // K panel: q-pair rows for 8 columns
        const int kp = tid >> 3, n0 = (tid & 7) * 8;
        const float* s0 = Kb + (2 * kp) * 1024 + col0 + n0;
        float4 u0 = *(const float4*)(s0);
        float4 u1 = *(const float4*)(s0 + 4);
        float4 w0 = *(const float4*)(s0 + 1024);
        float4 w1 = *(const float4*)(s0 + 1028);
        stpair(Bs, PLD64, 2 * kp, n0 + 0, u0.x, w0.x);
        stpair(Bs, PLD64, 2 * kp, n0 + 1, u0.y, w0.y);
        stpair(Bs, PLD64, 2 * kp, n0 + 2, u0.z, w0.z);
        stpair(Bs, PLD64, 2 * kp, n0 + 3, u0.w, w0.w);
        stpair(Bs, PLD64, 2 * kp, n0 + 4, u1.x, w1.x);
        stpair(Bs, PLD64, 2 * kp, n0 + 5, u1.y, w1.y);
        stpair(Bs, PLD64, 2 * kp, n0 + 6, u1.z, w1.z);
        stpair(Bs, PLD64, 2 * kp, n0 + 7, u1.w, w1.w);
    }
    __syncthreads();

    v8f acc[2][2] = {};
    #pragma unroll
    for (int ks = 0; ks < 8; ++ks) {
        const int kb = ks * 4;
        v2f a0 = ldfrag2(As, lane, 32 * wm + 0,  kb, PLD64);
        v2f a1 = ldfrag2(As, lane, 32 * wm + 16, kb, PLD64);
        v2f b0 = ldfrag2(Bs, lane, 32 * wn + 0,  kb, PLD64);
        v2f b1 = ldfrag2(Bs, lane, 32 * wn + 16, kb, PLD64);
        acc[0][0] = wmma_f32(a0, b0, acc[0][0]);
        acc[0][1] = wmma_f32(a0, b1, acc[0][1]);
        acc[1][0] = wmma_f32(a1, b0, acc[1][0]);
        acc[1][1] = wmma_f32(a1, b1, acc[1][1]);
    }
    const int kh = lane >> 4, r15 = lane & 15;
    #pragma unroll
    for (int mi = 0; mi < 2; ++mi)
        #pragma unroll
        for (int ni = 0; ni < 2; ++ni) {
            const int nrow = row0 + 32 * wm + 16 * mi + 8 * kh;
            const int mcol = col0 + 32 * wn + 16 * ni + r15;
            #pragma unroll
            for (int i = 0; i < 8; ++i)
                S[(nrow + i) * 1024 + mcol] = acc[mi][ni][i];
        }
}

// ---------------------------------------------------------------------------
// Row softmax over 1024 columns, one block per row, row held in registers.
// ---------------------------------------------------------------------------
__global__ __launch_bounds__(128) void softmax_rows_kernel(float* __restrict__ S) {
    float* p = S + blockIdx.x * 1024;
    const int tid = threadIdx.x;
    __shared__ float red[128];
    float vals[8];
    float mx = -3.402823466e+38f;
    #pragma unroll
    for (int i = 0; i < 8; ++i) { vals[i] = p[tid + 128 * i]; mx = fmaxf(mx, vals[i]); }
    red[tid] = mx; __syncthreads();
    for (int s = 64; s > 0; s >>= 1) {
        if (tid < s) red[tid] = fmaxf(red[tid], red[tid + s]);
        __syncthreads();
    }
    mx = red[0]; __syncthreads();
    float sum = 0.0f;
    #pragma unroll
    for (int i = 0; i < 8; ++i) { vals[i] = __expf(vals[i] - mx); sum += vals[i]; }
    red[tid] = sum; __syncthreads();
    for (int s = 64; s > 0; s >>= 1) {
        if (tid < s) red[tid] += red[tid + s];
        __syncthreads();
    }
    const float inv = 1.0f / red[0];
    #pragma unroll
    for (int i = 0; i < 8; ++i) p[tid + 128 * i] = vals[i] * inv;
}

// ---------------------------------------------------------------------------
// Attention output (one batch): D[c,n] = sum_m V[c,m]*S[n,m];
// fused epilogue: d_out = gamma*D + y   (y already resides in d_out).
// ---------------------------------------------------------------------------
__global__ __launch_bounds__(128) void attnout_wmma_kernel(
    const float* __restrict__ Vb, const float* __restrict__ S,
    const float* __restrict__ gamma, float* __restrict__ OutY, int b) {
    __shared__ float As[8 * PLD64 * 2];   // [mpair][c][2]
    __shared__ float Bs[8 * PLD64 * 2];   // [mpair][n][2]
    const int tid = threadIdx.x, lane = tid & 31, wave = tid >> 5;
    const int wm = wave >> 1, wn = wave & 1;
    const int c0 = blockIdx.y * 64;
    const int n0 = blockIdx.x * 64;

    v8f acc[2][2] = {};
    const int ar = tid >> 1, ak8 = (tid & 1) * 8;

    for (int k0 = 0; k0 < 1024; k0 += 16) {
        __syncthreads();
        {   // V rows (c) along m: 8 consecutive m for one c
            const float* src = Vb + (c0 + ar) * 1024 + k0 + ak8;
            float4 v0 = *(const float4*)(src);
            float4 v1 = *(const float4*)(src + 4);
            stpair(As, PLD64, ak8 + 0, ar, v0.x, v0.y);
            stpair(As, PLD64, ak8 + 2, ar, v0.z, v0.w);
            stpair(As, PLD64, ak8 + 4, ar, v1.x, v1.y);
            stpair(As, PLD64, ak8 + 6, ar, v1.z, v1.w);
        }
        {   // S rows (n) along m
            const float* src = S + (n0 + ar) * 1024 + k0 + ak8;
            float4 v0 = *(const float4*)(src);
            float4 v1 = *(const float4*)(src + 4);
            stpair(Bs, PLD64, ak8 + 0, ar, v0.x, v0.y);
            stpair(Bs, PLD64, ak8 + 2, ar, v0.z, v0.w);
            stpair(Bs, PLD64, ak8 + 4, ar, v1.x, v1.y);
            stpair(Bs, PLD64, ak8 + 6, ar, v1.z, v1.w);
        }
        __syncthreads();
        #pragma unroll
        for (int ks = 0; ks < 4; ++ks) {
            const int kb = ks * 4;
            v2f a0 = ldfrag2(As, lane, 32 * wm + 0,  kb, PLD64);
            v2f a1 = ldfrag2(As, lane, 32 * wm + 16, kb, PLD64);
            v2f b0 = ldfrag2(Bs, lane, 32 * wn + 0,  kb, PLD64);
            v2f b1 = ldfrag2(Bs, lane, 32 * wn + 16, kb, PLD64);
            acc[0][0] = wmma_f32(a0, b0, acc[0][0]);
            acc[0][1] = wmma_f32(a0, b1, acc[0][1]);
            acc[1][0] = wmma_f32(a1, b0, acc[1][0]);
            acc[1][1] = wmma_f32(a1, b1, acc[1][1]);
        }
    }
    const float g = gamma[0];
    const int kh = lane >> 4, r15 = lane & 15;
    #pragma unroll
    for (int mi = 0; mi < 2; ++mi)
        #pragma unroll
        for (int ni = 0; ni < 2; ++ni) {
            const int n = n0 + 32 * wn + 16 * ni + r15;
            #pragma unroll
            for (int i = 0; i < 8; ++i) {
                const int c = c0 + 32 * wm + 16 * mi + 8 * kh + i;
                const int idx = ((b * 256 + c) << 10) + n;
                OutY[idx] = g * acc[mi][ni][i] + OutY[idx];   // residual: y in OutY
            }
        }
}

// ---------------------------------------------------------------------------
extern "C" void kernel_launch(void* const* d_in, const int* in_sizes, int n_in,
                              void* d_out, int out_size, void* d_ws, size_t ws_size,
                              hipStream_t stream) {
    const float* x    = (const float*)d_in[0];
    const float* t    = (const float*)d_in[1];
    const float* w_t1 = (const float*)d_in[2];
    const float* b_t1 = (const float*)d_in[3];
    const float* w_t2 = (const float*)d_in[4];
    const float* b_t2 = (const float*)d_in[5];
    const float* w_c1 = (const float*)d_in[6];
    const float* b_c1 = (const float*)d_in[7];
    const float* w_c2 = (const float*)d_in[8];
    const float* b_c2 = (const float*)d_in[9];
    const float* w_tr = (const float*)d_in[10];
    const float* b_tr = (const float*)d_in[11];
    const float* bn1g = (const float*)d_in[12];
    const float* bn1b = (const float*)d_in[13];
    const float* bn2g = (const float*)d_in[14];
    const float* bn2b = (const float*)d_in[15];
    const float* wq   = (const float*)d_in[16];
    const float* bq   = (const float*)d_in[17];
    const float* wk   = (const float*)d_in[18];
    const float* bk   = (const float*)d_in[19];
    const float* wv   = (const float*)d_in[20];
    const float* bv   = (const float*)d_in[21];
    const float* gam  = (const float*)d_in[22];

    float* ws     = (float*)d_ws;
    float* te     = ws;                      // 4096 floats
    float* stats1 = ws + 4096;               // 512  (sum[256], sumsq[256])
    float* stats2 = stats1 + 512;            // 512
    float* bufA   = ws + 8192;               // 4,194,304 (h1)
    float* bufB   = bufA + 4194304;          // 4,194,304 (h2)
    float* Q      = bufB + 4194304;          // 524,288   [B,N,32]
    float* Kt     = Q + 524288;              // 524,288   [B,32,N]
    float* V      = Kt + 524288;             // 4,194,304 [B,C,N]
    float* S      = V + 4194304;             // 1,048,576 (one batch of scores)
    float* yOut   = (float*)d_out;           // y lives here, then final output

    zero_kernel<<<4, 256, 0, stream>>>(stats1, 1024);   // stats1 + stats2
    time_mlp_kernel<<<1, 256, 0, stream>>>(t, w_t1, b_t1, w_t2, b_t2, te);

    // h1 = bn1(relu(conv1(x))) + te
    conv3x3_wmma_kernel<<<dim3(128, 4), 128, 0, stream>>>(x, w_c1, b_c1, bufA, stats1, 1);
    bn_apply_kernel<<<16384, 256, 0, stream>>>(bufA, stats1, bn1g, bn1b, te);

    // h2 = bn2(relu(conv2(h1)))
    conv3x3_wmma_kernel<<<dim3(128, 4), 128, 0, stream>>>(bufA, w_c2, b_c2, bufB, stats2, 1);
    bn_apply_kernel<<<16384, 256, 0, stream>>>(bufB, stats2, bn2g, bn2b, nullptr);

    // y = conv_tr(h2)  (no relu, no BN) -> d_out
    conv3x3_wmma_kernel<<<dim3(128, 4), 128, 0, stream>>>(bufB, w_tr, b_tr, yOut, nullptr, 0);

    // QKV 1x1 projections (biases fused)
    proj1x1_wmma_kernel<<<dim3(256, 1), 128, 0, stream>>>(yOut, wq, bq, Q, 32, 1);
    proj1x1_wmma_kernel<<<dim3(256, 1), 128, 0, stream>>>(yOut, wk, bk, Kt, 32, 0);
    proj1x1_wmma_kernel<<<dim3(256, 8), 128, 0, stream>>>(yOut, wv, bv, V, 256, 0);

    // attention per batch (reuses one 4MB score buffer)
    for (int b = 0; b < 16; ++b) {
        scores_wmma_kernel<<<dim3(16, 16), 128, 0, stream>>>(Q + b * 32768, Kt + b * 32768, S);
        softmax_rows_kernel<<<1024, 128, 0, stream>>>(S);
        attnout_wmma_kernel<<<dim3(16, 4), 128, 0, stream>>>(V + b * 262144, S, gam, yOut, b);
    }
}